// DeLanNet_inverse_55671366091081
// MI455X (gfx1250) — compile-verified
//
#include <hip/hip_runtime.h>
#include <hip/hip_bf16.h>

typedef __attribute__((ext_vector_type(2))) float v2f;
typedef __attribute__((ext_vector_type(8))) float v8f;

#define DOF   7
#define TRIL  21
#define HID   256
#define ROWS_PER_TILE 16
#define WAVES 8
#define TILES_PER_WAVE 4
#define ROWS_PER_WG (WAVES * ROWS_PER_TILE * TILES_PER_WAVE)   // 512

// LDS row strides in floats, padded to keep ds_load_b64 8B-aligned and bank-conflict free
#define W1_STR 10     // W1^T : [N=256][K=8 used]
#define W2_STR 258    // W2^T : [N][K=256]
#define WB_STR 50     // per-wave work tile: cols 0..15 = h staging, 16..22 diag, 23..43 tril

__device__ __forceinline__ v8f wmma_f32(v2f a, v2f b, v8f c) {
    // D = A(16x4 f32) * B(4x16 f32) + C(16x16 f32)
    return __builtin_amdgcn_wmma_f32_16x16x4_f32(false, a, false, b, (short)0, c, false, false);
}

__global__ void __launch_bounds__(256)
delan_inverse_kernel(const float* __restrict__ x,
                     const float* __restrict__ W1d, const float* __restrict__ b1d,
                     const float* __restrict__ W2d, const float* __restrict__ b2d,
                     const float* __restrict__ W1o, const float* __restrict__ b1o,
                     const float* __restrict__ W2o, const float* __restrict__ b2o,
                     float* __restrict__ out)
{
    __shared__ float sW1d[256 * W1_STR];      // 10240 B
    __shared__ float sW1o[256 * W1_STR];      // 10240 B
    __shared__ float sW2d[16  * W2_STR];      // 16512 B
    __shared__ float sW2o[32  * W2_STR];      // 33024 B
    __shared__ float sWB [WAVES * 16 * WB_STR]; // 25600 B   total ~93.4 KB (<320 KB/WGP)

    const int tid = threadIdx.x;

    // ---- Stage weights transposed [N][K] with zero padding ----
    for (int idx = tid; idx < 256 * 8; idx += 256) {
        int n = idx >> 3, k = idx & 7;
        sW1d[n * W1_STR + k] = (k < DOF) ? W1d[k * HID + n] : 0.0f;
        sW1o[n * W1_STR + k] = (k < DOF) ? W1o[k * HID + n] : 0.0f;
    }
    for (int idx = tid; idx < 16 * 256; idx += 256) {
        int n = idx >> 8, k = idx & 255;
        sW2d[n * W2_STR + k] = (n < DOF) ? W2d[k * DOF + n] : 0.0f;
    }
    for (int idx = tid; idx < 32 * 256; idx += 256) {
        int n = idx >> 8, k = idx & 255;
        sW2o[n * W2_STR + k] = (n < TRIL) ? W2o[k * TRIL + n] : 0.0f;
    }
    __syncthreads();

    const int wave = tid >> 5;
    const int lane = tid & 31;
    const int lh   = lane >> 4;   // lane half: selects K pair (A/B frag) / row group (C)
    const int ln   = lane & 15;   // column (B/C frags) / row (A frag)
    float* wb = &sWB[wave * 16 * WB_STR];

    for (int t = 0; t < TILES_PER_WAVE; ++t) {
        const long rowBase = (long)blockIdx.x * ROWS_PER_WG
                           + wave * (ROWS_PER_TILE * TILES_PER_WAVE) + t * ROWS_PER_TILE;

        // A fragments of q (16x8, col 7 zero-padded). A layout: lane<16 holds K={0,1},
        // lane>=16 holds K={2,3} of the 16x4 step.
        const float* xr = x + (size_t)(rowBase + ln) * (3 * DOF);
        v2f a0, a1;
        {
            int k0 = 2 * lh;          // 0 or 2
            a0.x = xr[k0];  a0.y = xr[k0 + 1];
            int k1 = 4 + 2 * lh;      // 4 or 6
            a1.x = xr[k1];  a1.y = (k1 + 1 < DOF) ? xr[k1 + 1] : 0.0f;  // col 7 -> 0
        }

        // ================= diagonal branch =================
        v8f cd = {};
        for (int nt = 0; nt < 16; ++nt) {
            const int N = nt * 16 + ln;
            v8f c = {};
            v2f bA = *(const v2f*)&sW1d[N * W1_STR + 2 * lh];        // K = {0,1}/{2,3}
            v2f bB = *(const v2f*)&sW1d[N * W1_STR + 4 + 2 * lh];    // K = {4,5}/{6,7}
            c = wmma_f32(a0, bA, c);
            c = wmma_f32(a1, bB, c);
            const float bias = b1d[N];
#pragma unroll
            for (int r = 0; r < 8; ++r) {
                float h = fmaxf(c[r] + bias, 0.0f);                  // bias + ReLU
                wb[(r + 8 * lh) * WB_STR + ln] = h;                  // C layout -> LDS
            }
            asm volatile("" ::: "memory");                           // keep ds store->load order
#pragma unroll
            for (int s2 = 0; s2 < 4; ++s2) {
                v2f af = *(const v2f*)&wb[ln * WB_STR + 4 * s2 + 2 * lh];
                v2f bf = *(const v2f*)&sW2d[ln * W2_STR + nt * 16 + 4 * s2 + 2 * lh];
                cd = wmma_f32(af, bf, cd);
            }
            asm volatile("" ::: "memory");
        }
        {   // stash diag (+b2d) into per-row epilogue slots, cols 16..22
            float b2 = (ln < DOF) ? b2d[ln] : 0.0f;
#pragma unroll
            for (int r = 0; r < 8; ++r)
                if (ln < DOF) wb[(r + 8 * lh) * WB_STR + 16 + ln] = cd[r] + b2;
        }

        // ================= off-diagonal branch =================
        v8f co0 = {}, co1 = {};
        for (int nt = 0; nt < 16; ++nt) {
            const int N = nt * 16 + ln;
            v8f c = {};
            v2f bA = *(const v2f*)&sW1o[N * W1_STR + 2 * lh];
            v2f bB = *(const v2f*)&sW1o[N * W1_STR + 4 + 2 * lh];
            c = wmma_f32(a0, bA, c);
            c = wmma_f32(a1, bB, c);
            const float bias = b1o[N];
#pragma unroll
            for (int r = 0; r < 8; ++r) {
                float h = fmaxf(c[r] + bias, 0.0f);
                wb[(r + 8 * lh) * WB_STR + ln] = h;
            }
            asm volatile("" ::: "memory");
#pragma unroll
            for (int s2 = 0; s2 < 4; ++s2) {
                v2f af  = *(const v2f*)&wb[ln * WB_STR + 4 * s2 + 2 * lh];
                v2f bf0 = *(const v2f*)&sW2o[ ln        * W2_STR + nt * 16 + 4 * s2 + 2 * lh];
                v2f bf1 = *(const v2f*)&sW2o[(16 + ln)  * W2_STR + nt * 16 + 4 * s2 + 2 * lh];
                co0 = wmma_f32(af, bf0, co0);
                co1 = wmma_f32(af, bf1, co1);
            }
            asm volatile("" ::: "memory");
        }
        {   // stash tril (+b2o) into cols 23..43
            float t0 = b2o[ln];
            float t1 = (ln < TRIL - 16) ? b2o[16 + ln] : 0.0f;
#pragma unroll
            for (int r = 0; r < 8; ++r) {
                int M = r + 8 * lh;
                wb[M * WB_STR + 23 + ln] = co0[r] + t0;
                if (ln < TRIL - 16) wb[M * WB_STR + 23 + 16 + ln] = co1[r] + t1;
            }
        }
        asm volatile("" ::: "memory");

        // ================= per-row epilogue: H = L L^T, tau = qDDot @ H =================
        if (lane < 16) {
            const long row = rowBase + lane;
            const float* wr = &wb[lane * WB_STR];
            float L[DOF][DOF];
#pragma unroll
            for (int i = 0; i < DOF; ++i) {
#pragma unroll
                for (int j = 0; j < DOF; ++j) L[i][j] = 0.0f;
                L[i][i] = wr[16 + i];
            }
            {
                int tt = 0;
#pragma unroll
                for (int i = 1; i < DOF; ++i)
#pragma unroll
                    for (int j = 0; j < i; ++j) L[i][j] = wr[23 + (tt++)];
            }
            float w[DOF];
            const float* xq = x + (size_t)row * (3 * DOF) + 2 * DOF;
#pragma unroll
            for (int i = 0; i < DOF; ++i) w[i] = xq[i];
            // v = L^T w  (L strictly lower+diag: i >= k)
            float v[DOF];
#pragma unroll
            for (int k = 0; k < DOF; ++k) {
                float s = 0.0f;
#pragma unroll
                for (int i = k; i < DOF; ++i) s += w[i] * L[i][k];
                v[k] = s;
            }
            // tau_j = sum_{k<=j} v_k L[j][k]
            float* op = out + (size_t)row * DOF;
#pragma unroll
            for (int j = 0; j < DOF; ++j) {
                float s = 0.0f;
#pragma unroll
                for (int k = 0; k <= j; ++k) s += v[k] * L[j][k];
                op[j] = s;
            }
        }
        asm volatile("" ::: "memory");  // wb reused next tile
    }
}

extern "C" void kernel_launch(void* const* d_in, const int* in_sizes, int n_in,
                              void* d_out, int out_size, void* d_ws, size_t ws_size,
                              hipStream_t stream) {
    const float* x   = (const float*)d_in[0];
    const float* W1d = (const float*)d_in[1];
    const float* b1d = (const float*)d_in[2];
    const float* W2d = (const float*)d_in[3];
    const float* b2d = (const float*)d_in[4];
    const float* W1o = (const float*)d_in[5];
    const float* b1o = (const float*)d_in[6];
    const float* W2o = (const float*)d_in[7];
    const float* b2o = (const float*)d_in[8];
    float* out = (float*)d_out;

    const int rows = in_sizes[0] / (3 * DOF);     // 524288
    const int grid = rows / ROWS_PER_WG;          // 1024
    delan_inverse_kernel<<<grid, 256, 0, stream>>>(x, W1d, b1d, W2d, b2d,
                                                   W1o, b1o, W2o, b2o, out);
}